// Controller_12996571038405
// MI455X (gfx1250) — compile-verified
//
#include <hip/hip_runtime.h>
#include <math.h>

typedef __attribute__((ext_vector_type(2))) float v2f;
typedef __attribute__((ext_vector_type(8))) float v8f;

#define HSZ 32      // LSTM hidden size
#define GSZ 128     // 4*HSZ gate rows
#define OSZ 64      // num ops

__device__ __forceinline__ unsigned rotl32(unsigned x, int r) {
    return (x << r) | (x >> (32 - r));
}

// JAX threefry2x32: 20 rounds, key-injection every 4.
__device__ __forceinline__ void threefry2x32(unsigned k0, unsigned k1,
                                             unsigned c0, unsigned c1,
                                             unsigned& o0, unsigned& o1) {
    unsigned ks0 = k0, ks1 = k1, ks2 = 0x1BD11BDAu ^ k0 ^ k1;
    unsigned x0 = c0 + ks0, x1 = c1 + ks1;
    const int rotA[4] = {13, 15, 26, 6};
    const int rotB[4] = {17, 29, 16, 24};
    unsigned ks[3] = {ks0, ks1, ks2};
#pragma unroll
    for (int i = 0; i < 5; ++i) {
#pragma unroll
        for (int j = 0; j < 4; ++j) {
            int r = (i & 1) ? rotB[j] : rotA[j];
            x0 += x1; x1 = rotl32(x1, r); x1 ^= x0;
        }
        x0 += ks[(i + 1) % 3];
        x1 += ks[(i + 2) % 3] + (unsigned)(i + 1);
    }
    o0 = x0; o1 = x1;
}

__device__ __forceinline__ float sigmoidf(float x) {
    return 1.0f / (1.0f + __expf(-x));
}

__global__ __launch_bounds__(256, 1)
void controller_lstm_kernel(const float* __restrict__ input_vars,
                            const float* __restrict__ W_ih0, const float* __restrict__ W_hh0,
                            const float* __restrict__ b_ih0, const float* __restrict__ b_hh0,
                            const float* __restrict__ W_ih1, const float* __restrict__ W_hh1,
                            const float* __restrict__ b_ih1, const float* __restrict__ b_hh1,
                            const float* __restrict__ embd,
                            const float* __restrict__ W_pred, const float* __restrict__ b_pred,
                            const int* __restrict__ num_edge_p,
                            float* __restrict__ out, int out_size) {
    __shared__ float s_embd[OSZ * HSZ];
    __shared__ float s_b0[GSZ], s_b1[GSZ], s_bp[OSZ];
    __shared__ float s_vin[2 * HSZ];          // concatenated [x_or_h; h] input vector
    __shared__ float s_gates[GSZ];
    __shared__ float s_logits[OSZ];
    __shared__ float s_h0[HSZ], s_c0[HSZ], s_h1[HSZ], s_c1[HSZ];

    const int tid  = threadIdx.x;
    const int lane = tid & 31;
    const int wid  = tid >> 5;
    const int m    = lane & 15;   // row within 16-row tile
    const int half = lane >> 4;   // 0: K pair {0,1}; 1: K pair {2,3}

    // ---------------- init: zero output, stage embeddings/biases/state -------
    for (int i = tid; i < out_size; i += 256) out[i] = 0.0f;
    for (int i = tid; i < OSZ * HSZ; i += 256) s_embd[i] = embd[i];
    for (int i = tid; i < GSZ; i += 256) {
        s_b0[i] = b_ih0[i] + b_hh0[i];
        s_b1[i] = b_ih1[i] + b_hh1[i];
    }
    for (int i = tid; i < OSZ; i += 256) s_bp[i] = b_pred[i];
    if (tid < HSZ) {
        float xv = input_vars[tid];
        s_h0[tid] = 0.f; s_c0[tid] = 0.f; s_h1[tid] = 0.f; s_c1[tid] = 0.f;
        s_vin[tid] = xv;            // x0
        s_vin[HSZ + tid] = 0.f;     // h0 = 0
    }

    // ------------- preload WMMA A-operands (weights) into registers ----------
    // A-matrix 16x4 f32 layout: VGPR0 = K{0|2}, VGPR1 = K{1|3}; lanes 0-15 M,
    // lanes 16-31 M with K+2.  Wave w owns rows [16w, 16w+16).
    v2f a0[16], a1[16], ap[8];
    {
        const int row = 16 * wid + m;
        const int prow = 16 * (wid & 3) + m;
#pragma unroll
        for (int k = 0; k < 8; ++k) {
            const int col = 4 * k + 2 * half;
            a0[k].x     = W_ih0[row * HSZ + col];
            a0[k].y     = W_ih0[row * HSZ + col + 1];
            a0[k + 8].x = W_hh0[row * HSZ + col];
            a0[k + 8].y = W_hh0[row * HSZ + col + 1];
            a1[k].x     = W_ih1[row * HSZ + col];
            a1[k].y     = W_ih1[row * HSZ + col + 1];
            a1[k + 8].x = W_hh1[row * HSZ + col];
            a1[k + 8].y = W_hh1[row * HSZ + col + 1];
            ap[k].x     = W_pred[prow * HSZ + col];
            ap[k].y     = W_pred[prow * HSZ + col + 1];
        }
    }

    const int NE = num_edge_p[0];
    unsigned key0 = 0u, key1 = 42u;   // jax.random.key(42)
    float sum_lp = 0.0f, sum_ent = 0.0f;

    __syncthreads();

    for (int step = 0; step < NE; ++step) {
        // ============ cell 0 gates: gates = [W_ih0|W_hh0] @ [x;h0] ==========
        {
            v8f acc = {0.f, 0.f, 0.f, 0.f, 0.f, 0.f, 0.f, 0.f};
#pragma unroll
            for (int k = 0; k < 16; ++k) {
                const int col = 4 * k + 2 * half;
                v2f b; b.x = s_vin[col]; b.y = s_vin[col + 1];  // broadcast cols
                acc = __builtin_amdgcn_wmma_f32_16x16x4_f32(
                    false, a0[k], false, b, (short)0, acc, false, false);
            }
            if (m == 0) {
                const int base = 16 * wid + half * 8;
#pragma unroll
                for (int r = 0; r < 8; ++r) s_gates[base + r] = acc[r];
            }
        }
        __syncthreads();

        // ---- cell 0 elementwise update (wave 0) ----
        if (wid == 0) {
            float gi = s_gates[lane]           + s_b0[lane];
            float gf = s_gates[HSZ + lane]     + s_b0[HSZ + lane];
            float gg = s_gates[2 * HSZ + lane] + s_b0[2 * HSZ + lane];
            float go = s_gates[3 * HSZ + lane] + s_b0[3 * HSZ + lane];
            float c = sigmoidf(gf) * s_c0[lane] + sigmoidf(gi) * tanhf(gg);
            float h = sigmoidf(go) * tanhf(c);
            s_c0[lane] = c; s_h0[lane] = h;
            s_vin[lane] = h;                 // cell1 input = h0_new
            s_vin[HSZ + lane] = s_h1[lane];  // cell1 hidden = h1_old
        }
        __syncthreads();

        // ============ cell 1 gates ==========
        {
            v8f acc = {0.f, 0.f, 0.f, 0.f, 0.f, 0.f, 0.f, 0.f};
#pragma unroll
            for (int k = 0; k < 16; ++k) {
                const int col = 4 * k + 2 * half;
                v2f b; b.x = s_vin[col]; b.y = s_vin[col + 1];
                acc = __builtin_amdgcn_wmma_f32_16x16x4_f32(
                    false, a1[k], false, b, (short)0, acc, false, false);
            }
            if (m == 0) {
                const int base = 16 * wid + half * 8;
#pragma unroll
                for (int r = 0; r < 8; ++r) s_gates[base + r] = acc[r];
            }
        }
        __syncthreads();

        // ---- cell 1 elementwise update (wave 0) ----
        if (wid == 0) {
            float gi = s_gates[lane]           + s_b1[lane];
            float gf = s_gates[HSZ + lane]     + s_b1[HSZ + lane];
            float gg = s_gates[2 * HSZ + lane] + s_b1[2 * HSZ + lane];
            float go = s_gates[3 * HSZ + lane] + s_b1[3 * HSZ + lane];
            float c = sigmoidf(gf) * s_c1[lane] + sigmoidf(gi) * tanhf(gg);
            float h = sigmoidf(go) * tanhf(c);
            s_c1[lane] = c; s_h1[lane] = h;
        }
        __syncthreads();

        // ============ prediction head: logits = W_pred @ h1 + b (waves 0-3) =
        if (wid < 4) {
            v8f acc = {0.f, 0.f, 0.f, 0.f, 0.f, 0.f, 0.f, 0.f};
#pragma unroll
            for (int k = 0; k < 8; ++k) {
                const int col = 4 * k + 2 * half;
                v2f b; b.x = s_h1[col]; b.y = s_h1[col + 1];
                acc = __builtin_amdgcn_wmma_f32_16x16x4_f32(
                    false, ap[k], false, b, (short)0, acc, false, false);
            }
            if (m == 0) {
                const int base = 16 * wid + half * 8;
#pragma unroll
                for (int r = 0; r < 8; ++r)
                    s_logits[base + r] = acc[r] + s_bp[base + r];
            }
        }
        __syncthreads();

        // ============ sampling + log-softmax + entropy (wave 0) ============
        if (wid == 0) {
            // tanh-squashed logits; lane handles indices {lane, lane+32}
            float l0 = 2.5f * tanhf(s_logits[lane] * 0.2f);
            float l1 = 2.5f * tanhf(s_logits[HSZ + lane] * 0.2f);

            // key, sub = jax.random.split(key): threefry over counters [0..3]
            unsigned n0, n1, sk0, sk1;
            threefry2x32(key0, key1, 0u, 2u, n0, sk0);
            threefry2x32(key0, key1, 1u, 3u, n1, sk1);
            key0 = n0; key1 = n1;

            // uniform bits for 64 draws: pairs (i, i+32)
            unsigned ub0, ub1;
            threefry2x32(sk0, sk1, (unsigned)lane, (unsigned)(lane + 32), ub0, ub1);
            const float tiny = 1.1754944e-38f;
            float u0 = __uint_as_float((ub0 >> 9) | 0x3f800000u) - 1.0f;
            float u1 = __uint_as_float((ub1 >> 9) | 0x3f800000u) - 1.0f;
            u0 = fmaxf(u0 * (1.0f - tiny) + tiny, tiny);
            u1 = fmaxf(u1 * (1.0f - tiny) + tiny, tiny);
            float z0 = l0 - logf(-logf(u0));   // logits + gumbel
            float z1 = l1 - logf(-logf(u1));

            // argmax over 64 (ties -> lowest index)
            float bv; int bi;
            if (z0 >= z1) { bv = z0; bi = lane; } else { bv = z1; bi = lane + 32; }
#pragma unroll
            for (int off = 16; off >= 1; off >>= 1) {
                float ov = __shfl_xor(bv, off, 32);
                int   oi = __shfl_xor(bi, off, 32);
                if (ov > bv || (ov == bv && oi < bi)) { bv = ov; bi = oi; }
            }
            const int idx = bi;

            // log-softmax over the 64 squashed logits
            float mx = fmaxf(l0, l1);
#pragma unroll
            for (int off = 16; off >= 1; off >>= 1)
                mx = fmaxf(mx, __shfl_xor(mx, off, 32));
            float se = __expf(l0 - mx) + __expf(l1 - mx);
#pragma unroll
            for (int off = 16; off >= 1; off >>= 1)
                se += __shfl_xor(se, off, 32);
            float lse = mx + logf(se);
            float p0 = l0 - lse, p1 = l1 - lse;

            // entropy = -sum(exp(logp)*logp)
            float es = __expf(p0) * p0 + __expf(p1) * p1;
#pragma unroll
            for (int off = 16; off >= 1; off >>= 1)
                es += __shfl_xor(es, off, 32);
            sum_ent += -es;

            // lp = logp[idx]: broadcast from owning lane
            float mine = (idx < 32) ? p0 : p1;
            float lp = __shfl(mine, idx & 31, 32);
            sum_lp += lp;

            if (lane == 0) ((int*)out)[2 + step] = idx;

            // next input: x = embd[idx]; vin = [x; h0]
            float xv = s_embd[idx * HSZ + lane];
            s_vin[lane] = xv;
            s_vin[HSZ + lane] = s_h0[lane];
        }
        __syncthreads();
    }

    if (tid == 0) {
        out[0] = sum_lp;
        out[1] = sum_ent;
    }
}

extern "C" void kernel_launch(void* const* d_in, const int* in_sizes, int n_in,
                              void* d_out, int out_size, void* d_ws, size_t ws_size,
                              hipStream_t stream) {
    (void)in_sizes; (void)n_in; (void)d_ws; (void)ws_size;
    controller_lstm_kernel<<<1, 256, 0, stream>>>(
        (const float*)d_in[0],                      // input_vars
        (const float*)d_in[1], (const float*)d_in[2],   // W_ih0, W_hh0
        (const float*)d_in[3], (const float*)d_in[4],   // b_ih0, b_hh0
        (const float*)d_in[5], (const float*)d_in[6],   // W_ih1, W_hh1
        (const float*)d_in[7], (const float*)d_in[8],   // b_ih1, b_hh1
        (const float*)d_in[9],                      // embd
        (const float*)d_in[10], (const float*)d_in[11], // W_pred, b_pred
        (const int*)d_in[12],                       // num_edge
        (float*)d_out, out_size);
}